// GCNConv_6846177869851
// MI455X (gfx1250) — compile-verified
//
#include <hip/hip_runtime.h>
#include <hip/hip_bf16.h>

typedef __attribute__((ext_vector_type(2))) float v2f;
typedef __attribute__((ext_vector_type(8))) float v8f;

#define DFEAT 1024

// ---------------------------------------------------------------------------
// deg[i] = 1 (self loop) + #occurrences of i in row indices
// ---------------------------------------------------------------------------
__global__ void deg_init_kernel(int* __restrict__ deg, int n) {
  int i = blockIdx.x * blockDim.x + threadIdx.x;
  if (i < n) deg[i] = 1;
}

__global__ void deg_count_kernel(const int* __restrict__ ei, int* __restrict__ deg, int ne) {
  int e = blockIdx.x * blockDim.x + threadIdx.x;
  if (e < ne) atomicAdd(&deg[ei[e]], 1);   // ei[0][e] = row
}

__global__ void dinv_kernel(const int* __restrict__ deg, float* __restrict__ dis, int n) {
  int i = blockIdx.x * blockDim.x + threadIdx.x;
  if (i < n) dis[i] = rsqrtf((float)deg[i]);
}

// ---------------------------------------------------------------------------
// Exclusive scan of per-node edge counts (deg-1) -> offs, cursor.
// Single block of 1024 threads, LDS Hillis-Steele with running carry.
// n=20000 -> 20 chunks; negligible runtime.
// ---------------------------------------------------------------------------
__global__ void __launch_bounds__(1024) scan_kernel(const int* __restrict__ deg,
                                                    int* __restrict__ offs,
                                                    int* __restrict__ cursor, int n) {
  __shared__ int buf[1024];
  __shared__ int carry;
  const int tid = threadIdx.x;
  if (tid == 0) carry = 0;
  __syncthreads();
  for (int base = 0; base < n; base += 1024) {
    const int i = base + tid;
    const int v = (i < n) ? (deg[i] - 1) : 0;  // real (non-self) edges for node i
    buf[tid] = v;
    __syncthreads();
    for (int d = 1; d < 1024; d <<= 1) {
      const int t = (tid >= d) ? buf[tid - d] : 0;
      __syncthreads();
      buf[tid] += t;
      __syncthreads();
    }
    const int excl = buf[tid] - v + carry;
    if (i < n) { offs[i] = excl; cursor[i] = excl; }
    __syncthreads();
    if (tid == 0) carry += buf[1023];
    __syncthreads();
  }
}

// ---------------------------------------------------------------------------
// CSR fill: bucket each edge's col under its row. Only E cheap int atomics.
// ---------------------------------------------------------------------------
__global__ void fill_kernel(const int* __restrict__ ei, int* __restrict__ cursor,
                            int* __restrict__ cols, int ne) {
  int e = blockIdx.x * blockDim.x + threadIdx.x;
  if (e < ne) {
    const int r = ei[e];
    const int pos = atomicAdd(&cursor[r], 1);
    cols[pos] = ei[ne + e];
  }
}

// ---------------------------------------------------------------------------
// h = x @ W^T + b using V_WMMA_F32_16X16X4_F32 (exact fp32 path).
// One wave owns a 32x64 output strip: 8 accumulators; B fragments reused
// across both 16-row halves, A fragments reused across 4 column tiles.
// 6 loads : 8 WMMAs per K=4 step (~10.7 FLOP/B from cache).
// ---------------------------------------------------------------------------
__global__ void __launch_bounds__(256) gemm_wmma_kernel(
    const float* __restrict__ X, const float* __restrict__ Wm,
    const float* __restrict__ bias, float* __restrict__ H, int nrows) {
  const int gwave = (blockIdx.x * 256 + (int)threadIdx.x) >> 5;
  const int lane  = threadIdx.x & 31;
  const int ntiles = DFEAT / 64;            // 16 column strips of 64
  const int mTile = gwave / ntiles;
  const int nTile = gwave % ntiles;
  const int mBase = mTile * 32;
  if (mBase >= nrows) return;
  const int half = lane >> 4;
  const int l    = lane & 15;
  const int nBase = nTile * 64;

  const int r0 = min(mBase + l, nrows - 1);
  const int r1 = min(mBase + 16 + l, nrows - 1);
  const float* x0 = X + (size_t)r0 * DFEAT;
  const float* x1 = X + (size_t)r1 * DFEAT;
  const float* w0 = Wm + (size_t)(nBase + 0 * 16 + l) * DFEAT;
  const float* w1 = Wm + (size_t)(nBase + 1 * 16 + l) * DFEAT;
  const float* w2 = Wm + (size_t)(nBase + 2 * 16 + l) * DFEAT;
  const float* w3 = Wm + (size_t)(nBase + 3 * 16 + l) * DFEAT;

  v8f acc[2][4];
#pragma unroll
  for (int g = 0; g < 2; ++g)
#pragma unroll
    for (int t = 0; t < 4; ++t) acc[g][t] = (v8f){};

  for (int k0 = 0; k0 < DFEAT; k0 += 4) {
    const int kk = k0 + 2 * half;
    v2f a0 = *(const v2f*)(x0 + kk);
    v2f a1 = *(const v2f*)(x1 + kk);
    v2f b0 = *(const v2f*)(w0 + kk);
    v2f b1 = *(const v2f*)(w1 + kk);
    v2f b2 = *(const v2f*)(w2 + kk);
    v2f b3 = *(const v2f*)(w3 + kk);
    acc[0][0] = __builtin_amdgcn_wmma_f32_16x16x4_f32(false, a0, false, b0, (short)0, acc[0][0], false, false);
    acc[1][0] = __builtin_amdgcn_wmma_f32_16x16x4_f32(false, a1, false, b0, (short)0, acc[1][0], false, false);
    acc[0][1] = __builtin_amdgcn_wmma_f32_16x16x4_f32(false, a0, false, b1, (short)0, acc[0][1], false, false);
    acc[1][1] = __builtin_amdgcn_wmma_f32_16x16x4_f32(false, a1, false, b1, (short)0, acc[1][1], false, false);
    acc[0][2] = __builtin_amdgcn_wmma_f32_16x16x4_f32(false, a0, false, b2, (short)0, acc[0][2], false, false);
    acc[1][2] = __builtin_amdgcn_wmma_f32_16x16x4_f32(false, a1, false, b2, (short)0, acc[1][2], false, false);
    acc[0][3] = __builtin_amdgcn_wmma_f32_16x16x4_f32(false, a0, false, b3, (short)0, acc[0][3], false, false);
    acc[1][3] = __builtin_amdgcn_wmma_f32_16x16x4_f32(false, a1, false, b3, (short)0, acc[1][3], false, false);
  }

#pragma unroll
  for (int t = 0; t < 4; ++t) {
    const int o = nBase + t * 16 + l;
    const float bv = bias[o];
#pragma unroll
    for (int g = 0; g < 2; ++g) {
#pragma unroll
      for (int r = 0; r < 8; ++r) {
        const int m = mBase + g * 16 + r + 8 * half;
        if (m < nrows) H[(size_t)m * DFEAT + o] = acc[g][t][r] + bv;
      }
    }
  }
}

// ---------------------------------------------------------------------------
// Per-node gather: out[i,:] = h[i,:]/deg[i] + sum_{e: row=i} dis_i*dis_c*h[c,:]
// One block (256 threads x float4) per node; out written exactly once,
// zero fp32 atomics. h/out both L2-resident (164 MB < 192 MB L2).
// ---------------------------------------------------------------------------
__global__ void gather_kernel(const int* __restrict__ offs, const int* __restrict__ deg,
                              const int* __restrict__ cols, const float* __restrict__ H,
                              const float* __restrict__ dis, float* __restrict__ out, int n) {
  const int node = blockIdx.x;
  const int t = threadIdx.x;
  const float di = dis[node];
  const int beg = offs[node];
  const int cnt = deg[node] - 1;

  const float4* hn = (const float4*)(H + (size_t)node * DFEAT);
  float4 v = hn[t];
  const float s = di * di;   // self-loop norm = 1/deg
  float4 acc = make_float4(v.x * s, v.y * s, v.z * s, v.w * s);

  for (int k = 0; k < cnt; ++k) {
    const int c = cols[beg + k];
    const float norm = di * dis[c];
    const float4* hc = (const float4*)(H + (size_t)c * DFEAT);
    float4 w = hc[t];
    acc.x += norm * w.x;
    acc.y += norm * w.y;
    acc.z += norm * w.z;
    acc.w += norm * w.w;
  }
  ((float4*)(out + (size_t)node * DFEAT))[t] = acc;
}

extern "C" void kernel_launch(void* const* d_in, const int* in_sizes, int n_in,
                              void* d_out, int out_size, void* d_ws, size_t ws_size,
                              hipStream_t stream) {
  const float* x    = (const float*)d_in[0];
  const int*   ei   = (const int*)d_in[1];
  const float* Wm   = (const float*)d_in[2];
  const float* bias = (const float*)d_in[3];
  float* out = (float*)d_out;

  const int n  = in_sizes[0] / DFEAT;   // 20000 nodes
  const int ne = in_sizes[1] / 2;       // 320000 edges

  // workspace layout: H | deg | dis | offs | cursor | cols
  char* ws = (char*)d_ws;
  float* H = (float*)ws;
  size_t off = (size_t)n * DFEAT * sizeof(float);
  int* deg = (int*)(ws + off);      off += (size_t)n * sizeof(int);
  float* dis = (float*)(ws + off);  off += (size_t)n * sizeof(float);
  int* offs = (int*)(ws + off);     off += (size_t)n * sizeof(int);
  int* cursor = (int*)(ws + off);   off += (size_t)n * sizeof(int);
  int* cols = (int*)(ws + off);

  deg_init_kernel<<<(n + 255) / 256, 256, 0, stream>>>(deg, n);
  deg_count_kernel<<<(ne + 255) / 256, 256, 0, stream>>>(ei, deg, ne);
  dinv_kernel<<<(n + 255) / 256, 256, 0, stream>>>(deg, dis, n);
  scan_kernel<<<1, 1024, 0, stream>>>(deg, offs, cursor, n);
  fill_kernel<<<(ne + 255) / 256, 256, 0, stream>>>(ei, cursor, cols, ne);

  const int mtiles = (n + 31) / 32;
  const int waves  = mtiles * (DFEAT / 64);
  const int blocks = (waves * 32 + 255) / 256;
  gemm_wmma_kernel<<<blocks, 256, 0, stream>>>(x, Wm, bias, H, n);

  gather_kernel<<<n, 256, 0, stream>>>(offs, deg, cols, H, dis, out, n);
}